// NeRFTrainer_14963666059931
// MI455X (gfx1250) — compile-verified
//
#include <hip/hip_runtime.h>
#include <math.h>

// NeRF hierarchical sampling: one wave32 per ray, single-wave workgroups
// (barriers become S_NOP on CDNA5), async global->LDS prefetch of `u`,
// warp-scan CDF, LDS binary-search inverse CDF, registerized bitonic-256
// with lane-major layout (j<8 sub-passes are free register CEs; only 120
// cross-lane shuffles total), rank-merge, coalesced NT output stores.

#define N_COARSE 128
#define N_FINE   256
#define N_ALL    384

#ifndef HAVE_ASYNC_LDS
#if defined(__has_builtin)
#if __has_builtin(__builtin_amdgcn_global_load_async_to_lds_b128)
#define HAVE_ASYNC_LDS 1
#endif
#endif
#endif
#ifndef HAVE_ASYNC_LDS
#define HAVE_ASYNC_LDS 0
#endif

#if defined(__has_builtin)
#if __has_builtin(__builtin_amdgcn_s_wait_asynccnt)
#define WAIT_ASYNC0() __builtin_amdgcn_s_wait_asynccnt(0)
#endif
#endif
#ifndef WAIT_ASYNC0
#define WAIT_ASYNC0() asm volatile("s_wait_asynccnt 0" ::: "memory")
#endif

// Builtin signature (from hipcc diagnostic): pointers to 16-byte int vectors.
typedef int v4i __attribute__((vector_size(4 * sizeof(int))));
typedef __attribute__((address_space(1))) v4i* global_v4i_ptr;
typedef __attribute__((address_space(3))) v4i* lds_v4i_ptr;

__global__ __launch_bounds__(32) void nerf_hier_sample_kernel(
    const float* __restrict__ ray_o,
    const float* __restrict__ ray_d,
    const float* __restrict__ t_rand,
    const float* __restrict__ weights,
    const float* __restrict__ u_in,
    float* __restrict__ out)
{
    __shared__ float zc[N_COARSE];                 // coarse z (sorted by construction)
    __shared__ float cdf[N_COARSE];                // cdf[0..126]
    __shared__ __align__(16) float zf[N_FINE];     // u staging, later sorted fine z
    __shared__ float zall[N_ALL];                  // merged sorted samples

    const int lane = threadIdx.x;                  // 0..31 (wave32)
    const int r    = blockIdx.x;                   // ray id

    // ---------------- kick off async copy of u[r, :] into LDS --------------
    // (overlaps with z_coarse + CDF phases below)
#if HAVE_ASYNC_LDS
    {
        global_v4i_ptr g = (global_v4i_ptr)(u_in + (size_t)r * N_FINE + lane * 4);
        lds_v4i_ptr    l = (lds_v4i_ptr)(&zf[lane * 4]);
        __builtin_amdgcn_global_load_async_to_lds_b128(g, l, 0, 0);
        __builtin_amdgcn_global_load_async_to_lds_b128(g + 32, l + 32, 0, 0);
    }
#else
    #pragma unroll
    for (int m = 0; m < 8; ++m)
        zf[lane + 32 * m] = __builtin_nontemporal_load(&u_in[(size_t)r * N_FINE + lane + 32 * m]);
#endif

    // Pull origin/direction early (broadcast loads, latency hidden by later work)
    const float o0 = ray_o[3 * (size_t)r + 0];
    const float o1 = ray_o[3 * (size_t)r + 1];
    const float o2 = ray_o[3 * (size_t)r + 2];
    const float d0 = ray_d[3 * (size_t)r + 0];
    const float d1 = ray_d[3 * (size_t)r + 1];
    const float d2 = ray_d[3 * (size_t)r + 2];

    // ---------------- Phase A: stratified coarse z (closed form) -----------
    // z_j = 2 + 4*j/127 ; lower/upper are midpoints => zc is sorted ascending.
    {
        const float K = 4.0f / 127.0f;
        #pragma unroll
        for (int q = 0; q < 4; ++q) {
            int   j  = lane + 32 * q;
            float lo = (j == 0)   ? 2.0f : 2.0f + ((float)j - 0.5f) * K;
            float hi = (j == 127) ? 6.0f : 2.0f + ((float)j + 0.5f) * K;
            float tr = __builtin_nontemporal_load(&t_rand[(size_t)r * N_COARSE + j]);
            zc[j] = lo + (hi - lo) * tr;
        }
    }

    // ---------------- Phase B: normalized CDF over weights[1..126] ---------
    {
        const float* wrow = weights + (size_t)r * N_COARSE + 1;
        int base = 4 * lane;                       // this lane owns i = base..base+3
        float v0 = (base + 0 < 126) ? (__builtin_nontemporal_load(&wrow[base + 0]) + 1e-5f) : 0.0f;
        float v1 = (base + 1 < 126) ? (__builtin_nontemporal_load(&wrow[base + 1]) + 1e-5f) : 0.0f;
        float v2 = (base + 2 < 126) ? (__builtin_nontemporal_load(&wrow[base + 2]) + 1e-5f) : 0.0f;
        float v3 = (base + 3 < 126) ? (__builtin_nontemporal_load(&wrow[base + 3]) + 1e-5f) : 0.0f;
        float p0 = v0, p1 = p0 + v1, p2 = p1 + v2, p3 = p2 + v3;

        // inclusive wave scan of per-lane sums (5 shuffle steps on wave32)
        float s = p3;
        #pragma unroll
        for (int d = 1; d < 32; d <<= 1) {
            float t = __shfl_up(s, (unsigned)d, 32);
            if (lane >= d) s += t;
        }
        float excl  = s - p3;
        float total = __shfl(s, 31, 32);
        float invS  = 1.0f / total;

        if (lane == 0) cdf[0] = 0.0f;
        float pp[4] = {p0, p1, p2, p3};
        #pragma unroll
        for (int k = 0; k < 4; ++k) {
            int i = base + k;
            if (i < 126) cdf[i + 1] = (excl + pp[k]) * invS;
        }
    }

    // retire async u copy, publish zc/cdf (S_NOP barrier for single-wave WG)
#if HAVE_ASYNC_LDS
    WAIT_ASYNC0();
#endif
    __syncthreads();

    // ---------------- Phase C: inverse-CDF sampling -> registers -----------
    // Slot assignment to (lane,t) is an arbitrary permutation of the 256
    // u values (a sort doesn't care), so keep fully coalesced staged reads.
    float v[8];
    #pragma unroll
    for (int t = 0; t < 8; ++t) {
        int   j  = lane + 32 * t;                  // coalesced LDS read pattern
        float uu = zf[j];
        // searchsorted right over cdf[0..126]: ind = #{cdf <= uu}
        int lo = 0, hi = 127;
        while (lo < hi) {
            int mid = (lo + hi) >> 1;
            if (cdf[mid] <= uu) lo = mid + 1; else hi = mid;
        }
        int below = lo - 1; if (below < 0)   below = 0;
        int above = lo;     if (above > 126) above = 126;
        float cb = cdf[below], ca = cdf[above];
        float bb = 0.5f * (zc[below] + zc[below + 1]);   // bins_b
        float ba = 0.5f * (zc[above] + zc[above + 1]);   // bins_a
        float den = ca - cb; den = (den < 1e-5f) ? 1.0f : den;
        float tt  = (uu - cb) / den;
        v[t] = bb + tt * (ba - bb);
    }

    // ---------------- Phase D: registerized bitonic sort of 256 ------------
    // Lane-major layout: sorted index i = (lane<<3)|t.
    //   j >= 8 : partner lane = lane ^ (j>>3), same t  -> shfl_xor (120 total)
    //   j <  8 : partner t = t ^ j, same lane          -> register min/max (free)
    #pragma unroll
    for (int k = 2; k <= N_FINE; k <<= 1) {
        // cross-lane sub-passes (j >= 8); only exist for k >= 16
        #pragma unroll
        for (int j = k >> 1; j >= 8; j >>= 1) {
            int jl = j >> 3;                           // lane xor mask
            #pragma unroll
            for (int t = 0; t < 8; ++t) {
                int   i     = (lane << 3) | t;
                float other = __shfl_xor(v[t], jl, 32);
                bool  up    = ((i & k) == 0);
                bool  lower = ((lane & jl) == 0);      // this side is i < partner
                float mn = fminf(v[t], other), mx = fmaxf(v[t], other);
                v[t] = (up == lower) ? mn : mx;
            }
        }
        // in-register sub-passes (j < 8)
        {
            int jstart = ((k >> 1) < 8) ? (k >> 1) : 4;
            #pragma unroll
            for (int j = jstart; j >= 1; j >>= 1) {
                #pragma unroll
                for (int t = 0; t < 8; ++t) {
                    if ((t & j) == 0) {
                        int  t2 = t ^ j;               // t < t2
                        bool up = (k >= 16) ? ((lane & (k >> 3)) == 0)
                                            : ((t & k) == 0);
                        float a = v[t], b = v[t2];
                        float mn = fminf(a, b), mx = fmaxf(a, b);
                        v[t]  = up ? mn : mx;
                        v[t2] = up ? mx : mn;
                    }
                }
            }
        }
    }

    // publish sorted fine samples to LDS: 8 consecutive floats per lane
    // (vectorizable to 2x ds_store_b128); overwrites staged u (same wave).
    __syncthreads();
    #pragma unroll
    for (int t = 0; t < 8; ++t) zf[(lane << 3) | t] = v[t];
    __syncthreads();

    // ---------------- Phase E: stable rank-merge (zc sorted U zf sorted) ---
    #pragma unroll
    for (int q = 0; q < 4; ++q) {
        int   i = lane + 32 * q;
        float w = zc[i];
        int lo = 0, hi = N_FINE;                 // lower_bound in zf: #{zf < w}
        while (lo < hi) { int mid = (lo + hi) >> 1; if (zf[mid] < w) lo = mid + 1; else hi = mid; }
        zall[i + lo] = w;
    }
    #pragma unroll
    for (int t = 0; t < 8; ++t) {
        int   i = (lane << 3) | t;               // sorted index of v[t]
        float w = v[t];
        int lo = 0, hi = N_COARSE;               // upper_bound in zc: #{zc <= w}
        while (lo < hi) { int mid = (lo + hi) >> 1; if (zc[mid] <= w) lo = mid + 1; else hi = mid; }
        zall[i + lo] = w;
    }
    __syncthreads();

    // ---------------- Phase F: points = o + d*z, coalesced NT stores -------
    float* orow = out + (size_t)r * (N_ALL * 3);
    for (int q = 0; q < 36; ++q) {
        int   t = lane + 32 * q;                 // 0..1151, contiguous per wave
        int   s = (unsigned)t / 3u;
        int   c = t - 3 * s;
        float z = zall[s];
        float oc = (c == 0) ? o0 : ((c == 1) ? o1 : o2);
        float dc = (c == 0) ? d0 : ((c == 1) ? d1 : d2);
        __builtin_nontemporal_store(fmaf(dc, z, oc), &orow[t]);
    }
}

extern "C" void kernel_launch(void* const* d_in, const int* in_sizes, int n_in,
                              void* d_out, int out_size, void* d_ws, size_t ws_size,
                              hipStream_t stream) {
    const float* ray_o   = (const float*)d_in[0];
    const float* ray_d   = (const float*)d_in[1];
    const float* t_rand  = (const float*)d_in[2];
    const float* weights = (const float*)d_in[3];
    const float* u_in    = (const float*)d_in[4];
    float*       out     = (float*)d_out;

    const int N = in_sizes[0] / 3;               // 65536 rays

    nerf_hier_sample_kernel<<<dim3(N), dim3(32), 0, stream>>>(
        ray_o, ray_d, t_rand, weights, u_in, out);
}